// BandedMultiheadedAttention_27719718928460
// MI455X (gfx1250) — compile-verified
//
#include <hip/hip_runtime.h>
#include <hip/hip_bf16.h>

// ---------------------------------------------------------------------------
// Problem constants (match the reference)
// ---------------------------------------------------------------------------
constexpr int Bc  = 2;
constexpr int Lc  = 2048;   // sequence length (power of two: >>11 / &2047)
constexpr int Dc  = 1024;   // d_model
constexpr int Hc  = 8;      // heads
constexpr int DHc = 128;    // head dim (power of two: >>7 / &127)
constexpr int KW  = 64;     // window width

typedef __attribute__((ext_vector_type(2))) float v2f;
typedef __attribute__((ext_vector_type(8))) float v8f;
typedef __attribute__((ext_vector_type(4))) int   v4i;

#define WMMA_F32(A, B, C) \
    __builtin_amdgcn_wmma_f32_16x16x4_f32(false, (A), false, (B), (short)0, (C), false, false)

// ---------------------------------------------------------------------------
// Async global->LDS copy (ASYNCcnt path), with a safe fallback.
// Probe round 2: builtin exists with signature
//   (int4 __device__* /*global*/, int4 __shared__* /*lds*/, imm offset, imm cpol)
// ---------------------------------------------------------------------------
#if __has_builtin(__builtin_amdgcn_global_load_async_to_lds_b128)
#define HAVE_ASYNC_LDS 1
#else
#define HAVE_ASYNC_LDS 0
#endif

__device__ __forceinline__ void copy_b128_to_lds(const float* g, float* l) {
#if HAVE_ASYNC_LDS
    __builtin_amdgcn_global_load_async_to_lds_b128(
        (__attribute__((address_space(1))) v4i*)(void*)g,
        (__attribute__((address_space(3))) v4i*)(void*)l, 0, 0);
#else
    *(float4*)l = *(const float4*)g;
#endif
}

__device__ __forceinline__ void wait_lds_copies() {
#if HAVE_ASYNC_LDS
#if __has_builtin(__builtin_amdgcn_s_wait_asynccnt)
    __builtin_amdgcn_s_wait_asynccnt(0);
#else
    asm volatile("s_wait_asynccnt 0" ::: "memory");
#endif
    asm volatile("" ::: "memory");
#endif
}

// ---------------------------------------------------------------------------
// Index helpers for the generic GEMM.
// ---------------------------------------------------------------------------
__device__ __forceinline__ size_t a_index(int amode, int n, int kidx) {
    if (amode == 0) return (size_t)n * Dc + kidx;
    int b = n >> 11, l = n & (Lc - 1);
    int h = kidx >> 7, e = kidx & (DHc - 1);
    return (((size_t)b * Hc + h) * Lc + l) * DHc + e;
}

__device__ __forceinline__ size_t o_index(int omode, int n, int m) {
    if (omode == 1) return (size_t)n * Dc + m;
    int b = n >> 11, l = n & (Lc - 1);
    int h = m >> 7, e = m & (DHc - 1);
    return (((size_t)b * Hc + h) * Lc + l) * DHc + e;
}

// ---------------------------------------------------------------------------
// Full-fp32 WMMA GEMM:  C[n][m] = sum_k A[n][k] * W[m][k] + bias[m]
// One wave / workgroup; 16(n) x 64(m) strip; one A fragment feeds four
// independent v_wmma_f32_16x16x4_f32 accumulators per K step.
// ---------------------------------------------------------------------------
template <int AMODE, int OMODE>
__global__ __launch_bounds__(32)
void gemm_xwt_wmma(const float* __restrict__ A, const float* __restrict__ W,
                   const float* __restrict__ bias, float* __restrict__ out) {
    const int lane = threadIdx.x;
    const int m0   = blockIdx.x * 64;
    const int n0   = blockIdx.y * 16;
    const int half = lane >> 4;
    const int lr   = lane & 15;
    const int koff = half * 2;

    const int arow = n0 + lr;
    const float* w0 = W + (size_t)(m0 + lr) * Dc;
    const float* w1 = w0 + (size_t)16 * Dc;
    const float* w2 = w0 + (size_t)32 * Dc;
    const float* w3 = w0 + (size_t)48 * Dc;

    v8f c0 = {}, c1 = {}, c2 = {}, c3 = {};

#pragma unroll 4
    for (int kk = 0; kk < Dc; kk += 4) {
        const int kidx = kk + koff;
        v2f a  = *(const v2f*)(A + a_index(AMODE, arow, kidx));
        v2f b0 = *(const v2f*)(w0 + kidx);
        v2f b1 = *(const v2f*)(w1 + kidx);
        v2f b2 = *(const v2f*)(w2 + kidx);
        v2f b3 = *(const v2f*)(w3 + kidx);
        c0 = WMMA_F32(a, b0, c0);
        c1 = WMMA_F32(a, b1, c1);
        c2 = WMMA_F32(a, b2, c2);
        c3 = WMMA_F32(a, b3, c3);
    }

#pragma unroll
    for (int r = 0; r < 8; ++r) {
        const int n = n0 + r + half * 8;
        const int m = m0 + lr;
        out[o_index(OMODE, n, m)]      = c0[r] + bias[m];
        out[o_index(OMODE, n, m + 16)] = c1[r] + bias[m + 16];
        out[o_index(OMODE, n, m + 32)] = c2[r] + bias[m + 32];
        out[o_index(OMODE, n, m + 48)] = c3[r] + bias[m + 48];
    }
}

// ---------------------------------------------------------------------------
// WMMA banded attention via lattice decomposition.
//
// Head h has dilation d = 1<<(h>>1). Queries with equal (l mod d) form a
// lattice; a tile = 16 lattice-consecutive queries t0..t0+15 of residue r.
// Their key windows cover lattice positions u in [t0-31, t0+64] -> 96 cols.
//   Phase 0: async-copy V_lat tile (96 x 128 fp32, 48KB) into LDS (overlaps S).
//   Phase 1: S[16x96] = Q[16x128] * K_lat^T      (6 wmma accum x 32 K-steps)
//   Phase 2: mask (0 <= col-row < 64, u in [0,Ld)) + exact softmax in LDS.
//   Phase 3: O[16x128] = P[16x96] * V_lat        (8 wmma accum x 24 K-steps)
//   Phase 4: scale by 1/sum (per-row shfl broadcast) and store [B,H,L,DH].
// One wave per block (EXEC all ones for WMMA; no cross-wave sync needed).
// ---------------------------------------------------------------------------
__global__ __launch_bounds__(32)
void banded_attn_wmma(const float* __restrict__ q, const float* __restrict__ k,
                      const float* __restrict__ v, float* __restrict__ o) {
    __shared__ float Sbuf[16 * 96];      // scores -> probabilities
    __shared__ float Vt[96 * DHc];       // V lattice tile

    const int lane = threadIdx.x;
    const int lr   = lane & 15;
    const int half = lane >> 4;
    const int koff = half * 2;

    const int bh     = blockIdx.x >> 7;          // / (Lc/16)
    const int within = blockIdx.x & 127;
    const int h      = bh & (Hc - 1);
    const int s      = h >> 1;                   // log2 dilation
    const int Ld     = Lc >> s;                  // lattice length
    const int lt     = 7 - s;                    // log2(tiles per residue)
    const int r      = within >> lt;             // residue class
    const int t0     = (within & ((1 << lt) - 1)) << 4;

    const float* qb = q + (size_t)bh * Lc * DHc;
    const float* kb = k + (size_t)bh * Lc * DHc;
    const float* vb = v + (size_t)bh * Lc * DHc;
    float*       ob = o + (size_t)bh * Lc * DHc;

    // ---- Phase 0: kick off the async V-lattice-tile DMA into LDS ----------
    for (int it = 0; it < 96; ++it) {
        const int u   = t0 - 31 + it;
        const int uc  = min(max(u, 0), Ld - 1);
        const int tok = r + (uc << s);
        copy_b128_to_lds(vb + (size_t)tok * DHc + lane * 4, &Vt[it * DHc + lane * 4]);
    }

    // ---- Phase 1: S = Q * K_lat^T ------------------------------------------
    const float* qrow = qb + (size_t)(r + ((t0 + lr) << s)) * DHc;
    const float* kp0; const float* kp1; const float* kp2;
    const float* kp3; const float* kp4; const float* kp5;
    {
        auto krow = [&](int nt) {
            const int u  = t0 - 31 + nt * 16 + lr;
            const int uc = min(max(u, 0), Ld - 1);
            return kb + (size_t)(r + (uc << s)) * DHc;
        };
        kp0 = krow(0); kp1 = krow(1); kp2 = krow(2);
        kp3 = krow(3); kp4 = krow(4); kp5 = krow(5);
    }

    v8f s0 = {}, s1 = {}, s2 = {}, s3 = {}, s4 = {}, s5 = {};
#pragma unroll 4
    for (int kk = 0; kk < DHc; kk += 4) {
        const int kx = kk + koff;
        v2f a  = *(const v2f*)(qrow + kx);
        v2f b0 = *(const v2f*)(kp0 + kx);
        v2f b1 = *(const v2f*)(kp1 + kx);
        v2f b2 = *(const v2f*)(kp2 + kx);
        v2f b3 = *(const v2f*)(kp3 + kx);
        v2f b4 = *(const v2f*)(kp4 + kx);
        v2f b5 = *(const v2f*)(kp5 + kx);
        s0 = WMMA_F32(a, b0, s0);
        s1 = WMMA_F32(a, b1, s1);
        s2 = WMMA_F32(a, b2, s2);
        s3 = WMMA_F32(a, b3, s3);
        s4 = WMMA_F32(a, b4, s4);
        s5 = WMMA_F32(a, b5, s5);
    }

    // ---- Phase 2a: masked store of S to LDS --------------------------------
    auto storeS = [&](const v8f& sv, int nt) {
        const int c = nt * 16 + lr;
        const int u = t0 - 31 + c;
#pragma unroll
        for (int rr = 0; rr < 8; ++rr) {
            const int row = rr + half * 8;
            const int j   = c - row;                 // window slot 0..63
            const bool ok = (u >= 0) & (u < Ld) & (j >= 0) & (j < KW);
            Sbuf[row * 96 + c] = ok ? sv[rr] : -1.0e30f;
        }
    };
    storeS(s0, 0); storeS(s1, 1); storeS(s2, 2);
    storeS(s3, 3); storeS(s4, 4); storeS(s5, 5);
    __syncthreads();

    // ---- Phase 2b: exact softmax (unnormalized exp kept in LDS) ------------
    // lane handles row=lr, columns [half*48, half*48+48)
    const int cbase = lr * 96 + half * 48;
    float mx = -3.0e38f;
#pragma unroll 8
    for (int c = 0; c < 48; ++c) mx = fmaxf(mx, Sbuf[cbase + c]);
    mx = fmaxf(mx, __shfl_xor(mx, 16));
    float sum = 0.f;
#pragma unroll 8
    for (int c = 0; c < 48; ++c) {
        const float e = __expf(Sbuf[cbase + c] - mx);
        Sbuf[cbase + c] = e;
        sum += e;
    }
    sum += __shfl_xor(sum, 16);
    const float inv = 1.0f / sum;
    __syncthreads();
    wait_lds_copies();   // V tile resident before the PV phase

    // ---- Phase 3: O = P * V_lat --------------------------------------------
    v8f o0 = {}, o1 = {}, o2 = {}, o3 = {}, o4 = {}, o5 = {}, o6 = {}, o7 = {};
#pragma unroll 2
    for (int kp = 0; kp < 96; kp += 4) {
        const int kx = kp + koff;
        v2f a = *(const v2f*)(&Sbuf[lr * 96 + kx]);
        auto bfrag = [&](int nv) {
            v2f b;
            b[0] = Vt[(kx)     * DHc + nv * 16 + lr];
            b[1] = Vt[(kx + 1) * DHc + nv * 16 + lr];
            return b;
        };
        o0 = WMMA_F32(a, bfrag(0), o0);
        o1 = WMMA_F32(a, bfrag(1), o1);
        o2 = WMMA_F32(a, bfrag(2), o2);
        o3 = WMMA_F32(a, bfrag(3), o3);
        o4 = WMMA_F32(a, bfrag(4), o4);
        o5 = WMMA_F32(a, bfrag(5), o5);
        o6 = WMMA_F32(a, bfrag(6), o6);
        o7 = WMMA_F32(a, bfrag(7), o7);
    }

    // ---- Phase 4: scale by 1/sum and store ---------------------------------
    float invr[8];
    size_t rowbase[8];
#pragma unroll
    for (int rr = 0; rr < 8; ++rr) {
        const int row = rr + half * 8;
        invr[rr] = __shfl(inv, row);                 // lanes 0..15 hold rows 0..15
        const int tok = r + ((t0 + row) << s);
        rowbase[rr] = (size_t)tok * DHc;
    }
    auto storeO = [&](const v8f& ov, int nv) {
#pragma unroll
        for (int rr = 0; rr < 8; ++rr)
            ob[rowbase[rr] + nv * 16 + lr] = ov[rr] * invr[rr];
    };
    storeO(o0, 0); storeO(o1, 1); storeO(o2, 2); storeO(o3, 3);
    storeO(o4, 4); storeO(o5, 5); storeO(o6, 6); storeO(o7, 7);
}

// ---------------------------------------------------------------------------
// Launcher. Input order: query,key,value,Wq,bq,Wk,bk,Wv,bv,Wc,bc (fp32).
// Workspace: q,k,v,attn in [B,H,L,DH] fp32 = 4 x 16 MB.
// ---------------------------------------------------------------------------
extern "C" void kernel_launch(void* const* d_in, const int* in_sizes, int n_in,
                              void* d_out, int out_size, void* d_ws, size_t ws_size,
                              hipStream_t stream) {
    const float* query = (const float*)d_in[0];
    const float* key_  = (const float*)d_in[1];
    const float* value = (const float*)d_in[2];
    const float* Wq    = (const float*)d_in[3];
    const float* bq    = (const float*)d_in[4];
    const float* Wk    = (const float*)d_in[5];
    const float* bk    = (const float*)d_in[6];
    const float* Wv    = (const float*)d_in[7];
    const float* bv    = (const float*)d_in[8];
    const float* Wc    = (const float*)d_in[9];
    const float* bc    = (const float*)d_in[10];
    float* out = (float*)d_out;

    const size_t per = (size_t)Bc * Hc * Lc * DHc;   // 4M floats
    float* qws = (float*)d_ws;
    float* kws = qws + per;
    float* vws = kws + per;
    float* aws = vws + per;

    dim3 ggrid(Dc / 64, (Bc * Lc) / 16);             // (16, 256)
    gemm_xwt_wmma<0, 0><<<ggrid, 32, 0, stream>>>(query, Wq, bq, qws);
    gemm_xwt_wmma<0, 0><<<ggrid, 32, 0, stream>>>(key_, Wk, bk, kws);
    gemm_xwt_wmma<0, 0><<<ggrid, 32, 0, stream>>>(value, Wv, bv, vws);

    // 2048 lattice tiles: 16 (b,h) x 128 tiles each, one wave per tile
    banded_attn_wmma<<<Bc * Hc * (Lc / 16), 32, 0, stream>>>(qws, kws, vws, aws);

    gemm_xwt_wmma<1, 1><<<ggrid, 32, 0, stream>>>(aws, Wc, bc, out);
}